// MERAnet_12979391169072
// MI455X (gfx1250) — compile-verified
//
#include <hip/hip_runtime.h>

// MERA/MPS hierarchical classifier for MI455X (gfx1250, wave32).
// i-stage (bond 16) mapped to V_WMMA_F32_16X16X4_F32: batch(16) x bond(16)
// state matrix V chained through 3 sites; two shared-B matmuls per site
// (V@A1, V@Ad) combined with per-batch scalar x_s. One wave per patch.

typedef float v2f __attribute__((ext_vector_type(2)));
typedef float v8f __attribute__((ext_vector_type(8)));

__global__ void copy_kernel(const float* __restrict__ src,
                            float* __restrict__ dst, int n) {
  int i = blockIdx.x * blockDim.x + threadIdx.x;
  if (i < n) dst[i] = src[i];
}

// ---------------- disentangler MPS: bond 4, out 4, in-place on inner region
__global__ __launch_bounds__(256)
void d_mps_kernel(float* __restrict__ img,          // (16,H,H) in/out
                  const float* __restrict__ A,      // (Pd,4,2,4,4)
                  const float* __restrict__ v,      // (Pd,4)
                  const float* __restrict__ w,      // (Pd,4,4)
                  int H, int nh) {
  int tid = blockIdx.x * blockDim.x + threadIdx.x;
  int Pd = nh * nh;
  if (tid >= Pd * 16) return;
  int p = tid >> 4, b = tid & 15;
  int pr = p / nh, pc = p % nh;
  int r0 = 1 + 2 * pr, c0 = 1 + 2 * pc;
  float* base = img + (size_t)b * H * H;

  float xs[4];
  xs[0] = base[r0 * H + c0];
  xs[1] = base[r0 * H + c0 + 1];
  xs[2] = base[(r0 + 1) * H + c0];
  xs[3] = base[(r0 + 1) * H + c0 + 1];

  const float* Ap = A + (size_t)p * 128;   // 4 sites * 2 feats * 16
  const float* vp = v + (size_t)p * 4;

  float vec[4];
  {
    const float* A0 = Ap;            // phi0 = x   -> f=0
    const float* A1 = Ap + 16;       // phi1 = 1-x -> f=1
    float x = xs[0];
    #pragma unroll
    for (int j = 0; j < 4; ++j) {
      float acc = 0.f;
      #pragma unroll
      for (int i = 0; i < 4; ++i) {
        float a1 = A1[i * 4 + j];
        acc += vp[i] * (a1 + x * (A0[i * 4 + j] - a1));
      }
      vec[j] = acc;
    }
  }
  #pragma unroll
  for (int s = 1; s < 4; ++s) {
    const float* A0 = Ap + s * 32;
    const float* A1 = A0 + 16;
    float x = xs[s];
    float nv[4];
    #pragma unroll
    for (int j = 0; j < 4; ++j) {
      float acc = 0.f;
      #pragma unroll
      for (int i = 0; i < 4; ++i) {
        float a1 = A1[i * 4 + j];
        acc += vec[i] * (a1 + x * (A0[i * 4 + j] - a1));
      }
      nv[j] = acc;
    }
    #pragma unroll
    for (int j = 0; j < 4; ++j) vec[j] = nv[j];
  }
  const float* wp = w + (size_t)p * 16;
  #pragma unroll
  for (int o = 0; o < 4; ++o) {
    float acc = 0.f;
    #pragma unroll
    for (int i = 0; i < 4; ++i) acc += vec[i] * wp[i * 4 + o];
    acc = acc > 0.f ? acc : 0.f;
    base[(r0 + (o >> 1)) * H + (c0 + (o & 1))] = acc;
  }
}

// ---------------- isometry MPS: bond 16, out 1; one wave per patch, WMMA f32
__global__ __launch_bounds__(32)
void i_mps_kernel(const float* __restrict__ img,    // (16,H,H)
                  float* __restrict__ out,          // (16,nI,nI)
                  const float* __restrict__ A,      // (Pi,4,2,16,16)
                  const float* __restrict__ v,      // (Pi,16)
                  const float* __restrict__ w,      // (Pi,16,1)
                  int H, int nI) {
  __shared__ float Vbuf[256];   // C-layout staging: [vgpr 0..7][lane 0..31]
  __shared__ float xbuf[64];    // [site][batch]
  __shared__ float ubuf[32];    // u1[0..15], ud[0..15]
  __shared__ float wbuf[16];

  const int lane = threadIdx.x;          // 0..31, one wave
  const int p    = blockIdx.x;
  const int Pi   = nI * nI;
  const int pi_  = p / nI, pj = p % nI;
  const int half = lane >> 4;            // lane half (WMMA layout split)
  const int n    = lane & 15;

  // gather per-patch pixel values for all 16 batches, 4 sites
  for (int t = lane; t < 64; t += 32) {
    int b = t >> 2, s = t & 3;
    xbuf[s * 16 + b] =
        img[(size_t)b * H * H + (2 * pi_ + (s >> 1)) * H + (2 * pj + (s & 1))];
  }
  if (lane < 16) wbuf[lane] = w[(size_t)p * 16 + lane];
  __syncthreads();

  const float* Ap = A + (size_t)p * 2048;    // 4 sites * 2 feats * 256
  const float* vp = v + (size_t)p * 16;

  // ---- site 0: V0[m,j] = (v@A1)[j] + x0[m]*(v@Ad)[j]
  {
    const float* A0 = Ap;          // f=0
    const float* A1 = Ap + 256;    // f=1
    float acc = 0.f;
    if (half == 0) {
      for (int i = 0; i < 16; ++i) acc += vp[i] * A1[i * 16 + n];
    } else {
      for (int i = 0; i < 16; ++i)
        acc += vp[i] * (A0[i * 16 + n] - A1[i * 16 + n]);
    }
    ubuf[lane] = acc;
  }
  __syncthreads();

  float Vc[8];                     // C/D layout: VGPR i <-> batch row i+8*half
  {
    float u1n = ubuf[n], udn = ubuf[16 + n];
    #pragma unroll
    for (int i = 0; i < 8; ++i) {
      int m = i + 8 * half;
      Vc[i] = u1n + xbuf[m] * udn;
    }
  }

  // A-operand fetch base: element (M=n, K) lives at
  // Vbuf[(n&7)*32 + 16*(n>>3) + K]; this lane needs K = 4c + 2*half (+1)
  const int abase = (n & 7) * 32 + ((n >> 3) << 4) + 2 * half;

  for (int s = 1; s < 4; ++s) {
    __syncthreads();                       // prior readers of Vbuf done
    #pragma unroll
    for (int i = 0; i < 8; ++i) Vbuf[i * 32 + lane] = Vc[i];
    __syncthreads();

    v2f a[4];
    #pragma unroll
    for (int c = 0; c < 4; ++c) {
      a[c].x = Vbuf[abase + 4 * c];
      a[c].y = Vbuf[abase + 4 * c + 1];
    }

    const float* A0 = Ap + s * 512;
    const float* A1 = A0 + 256;
    v8f P = {};      // V @ A1
    v8f Q = {};      // V @ (A0 - A1)
    #pragma unroll
    for (int c = 0; c < 4; ++c) {
      int k0 = 4 * c + 2 * half;           // B rows for this lane half
      float a1a = A1[k0 * 16 + n];
      float a1b = A1[(k0 + 1) * 16 + n];
      v2f b1, bd;
      b1.x = a1a;
      b1.y = a1b;
      bd.x = A0[k0 * 16 + n] - a1a;
      bd.y = A0[(k0 + 1) * 16 + n] - a1b;
      P = __builtin_amdgcn_wmma_f32_16x16x4_f32(false, a[c], false, b1,
                                                (short)0, P, false, false);
      Q = __builtin_amdgcn_wmma_f32_16x16x4_f32(false, a[c], false, bd,
                                                (short)0, Q, false, false);
    }
    #pragma unroll
    for (int i = 0; i < 8; ++i) {
      int m = i + 8 * half;
      Vc[i] = P[i] + xbuf[s * 16 + m] * Q[i];
    }
  }

  // ---- output: o[b] = relu(V3[b,:] . w)
  __syncthreads();
  #pragma unroll
  for (int i = 0; i < 8; ++i) Vbuf[i * 32 + lane] = Vc[i];
  __syncthreads();
  if (lane < 16) {
    const float* row = &Vbuf[(lane & 7) * 32 + ((lane >> 3) << 4)];
    float acc = 0.f;
    for (int nn = 0; nn < 16; ++nn) acc += row[nn] * wbuf[nn];
    acc = acc > 0.f ? acc : 0.f;
    out[(size_t)lane * Pi + p] = acc;
  }
}

// ---------------- final MPS: 1 patch, bond 16, out 10; one thread per batch
__global__ __launch_bounds__(32)
void final_mps_kernel(const float* __restrict__ hin,  // (16,2,2) flat
                      float* __restrict__ outp,       // (16,10)
                      const float* __restrict__ A,    // (1,4,2,16,16)
                      const float* __restrict__ v,    // (1,16)
                      const float* __restrict__ w) {  // (1,16,10)
  int b = threadIdx.x;
  if (b >= 16) return;
  float xs[4];
  #pragma unroll
  for (int s = 0; s < 4; ++s) xs[s] = hin[b * 4 + s];
  float vec[16];
  {
    const float* A0 = A;
    const float* A1 = A + 256;
    float x = xs[0];
    for (int j = 0; j < 16; ++j) {
      float acc = 0.f;
      for (int i = 0; i < 16; ++i) {
        float a1 = A1[i * 16 + j];
        acc += v[i] * (a1 + x * (A0[i * 16 + j] - a1));
      }
      vec[j] = acc;
    }
  }
  for (int s = 1; s < 4; ++s) {
    const float* A0 = A + s * 512;
    const float* A1 = A0 + 256;
    float x = xs[s];
    float nv[16];
    for (int j = 0; j < 16; ++j) {
      float acc = 0.f;
      for (int i = 0; i < 16; ++i) {
        float a1 = A1[i * 16 + j];
        acc += vec[i] * (a1 + x * (A0[i * 16 + j] - a1));
      }
      nv[j] = acc;
    }
    for (int j = 0; j < 16; ++j) vec[j] = nv[j];
  }
  for (int o = 0; o < 10; ++o) {
    float acc = 0.f;
    for (int i = 0; i < 16; ++i) acc += vec[i] * w[i * 10 + o];
    outp[b * 10 + o] = acc > 0.f ? acc : 0.f;
  }
}

extern "C" void kernel_launch(void* const* d_in, const int* in_sizes, int n_in,
                              void* d_out, int out_size, void* d_ws,
                              size_t ws_size, hipStream_t stream) {
  (void)in_sizes; (void)n_in; (void)out_size; (void)ws_size;
  const float* x = (const float*)d_in[0];

  float* buf0 = (float*)d_ws;                 // 16*128*128 f32 = 1 MB
  float* buf1 = buf0 + 16 * 128 * 128;        // 1 MB

  // stage input image into buf0
  {
    int nelem = 16 * 128 * 128;
    copy_kernel<<<(nelem + 255) / 256, 256, 0, stream>>>(x, buf0, nelem);
  }

  float* cur = buf0;
  float* nxt = buf1;
  int H = 128;
  for (int l = 0; l < 6; ++l) {
    const int base = 1 + l * 6;
    const float* dA = (const float*)d_in[base + 0];
    const float* dv = (const float*)d_in[base + 1];
    const float* dw = (const float*)d_in[base + 2];
    const float* iA = (const float*)d_in[base + 3];
    const float* iv = (const float*)d_in[base + 4];
    const float* iw = (const float*)d_in[base + 5];

    int nh = (H - 2) / 2;
    int Pd = nh * nh;
    d_mps_kernel<<<(Pd * 16 + 255) / 256, 256, 0, stream>>>(cur, dA, dv, dw, H,
                                                            nh);

    int nI = H / 2;
    int Pi = nI * nI;
    i_mps_kernel<<<Pi, 32, 0, stream>>>(cur, nxt, iA, iv, iw, H, nI);

    float* t = cur; cur = nxt; nxt = t;
    H /= 2;
  }

  const float* fA = (const float*)d_in[37];
  const float* fv = (const float*)d_in[38];
  const float* fw = (const float*)d_in[39];
  final_mps_kernel<<<1, 32, 0, stream>>>(cur, (float*)d_out, fA, fv, fw);
}